// DeepHarmony_76562087019103
// MI455X (gfx1250) — compile-verified
//
#include <hip/hip_runtime.h>
#include <hip/hip_bf16.h>

// ---------------------------------------------------------------------------
// DeepHarmony (DeepSpeech-like) pipeline for MI455X / gfx1250, f32 WMMA path.
//   conv0 -> 3x residual(LN+GELU+conv3x3 twice, +res) -> BN+FC1 (WMMA GEMM)
//   -> 5x LayerNorm-LSTM (igates batched WMMA GEMM + persistent-WGP recurrence
//      with register-resident Whh WMMA fragments) -> BN+FC2 head.
// Matrix math: V_WMMA_F32_16X16X4_F32 (exact f32, matches reference numerics).
// Bulk LDS staging: GLOBAL_LOAD_ASYNC_TO_LDS_B128 + S_WAIT_ASYNCCNT.
// ---------------------------------------------------------------------------

typedef __attribute__((ext_vector_type(2))) float v2f;
typedef __attribute__((ext_vector_type(8))) float v8f;

__device__ __forceinline__ v8f wmma_f32(v2f a, v2f b, v8f c) {
#if defined(__gfx1250__)
  // D = A(16x4 f32) * B(4x16 f32) + C(16x16 f32)
  return __builtin_amdgcn_wmma_f32_16x16x4_f32(false, a, false, b, (short)0, c,
                                               false, false);
#else
  (void)a; (void)b;
  return c;
#endif
}

// Async copy of 16 bytes: global -> LDS, tracked by ASYNCcnt.
// VDST VGPR = 32-bit LDS byte offset (low 32 bits of generic shared pointer);
// VADDR = 64-bit global address (GV mode, saddr = off).
__device__ __forceinline__ void async_copy16(void* lds_dst, const void* gsrc) {
#if defined(__gfx1250__)
  unsigned int loff = (unsigned int)(unsigned long long)lds_dst;
  asm volatile("global_load_async_to_lds_b128 %0, %1, off"
               :
               : "v"(loff), "v"(gsrc)
               : "memory");
#else
  (void)lds_dst; (void)gsrc;
#endif
}

__device__ __forceinline__ void wait_async0() {
#if defined(__gfx1250__)
  asm volatile("s_wait_asynccnt 0x0" ::: "memory");
#endif
}

__device__ __forceinline__ float sigf(float x) { return 1.0f / (1.0f + expf(-x)); }
__device__ __forceinline__ float geluf(float x) {
  return 0.5f * x * (1.0f + erff(x * 0.70710678118654752f));
}

// Problem constants
#define NB   16     // batch
#define F0   129
#define T0   2048
#define CC   32     // conv channels
#define F1   65
#define T1   1024
#define CF   2080   // CC*F1
#define RD   128    // rnn dim
#define HH   128    // hidden
#define GG   512    // 4*H
#define NCLS 26
#define NROWS (T1*NB)   // 16384 sequence rows

// ---------------------------------------------------------------------------
// conv0: 1->32 ch, 3x3, stride 2, pad 1.  x[16,1,129,2048] -> out[16,32,65,1024]
// ---------------------------------------------------------------------------
__global__ void conv0_kernel(const float* __restrict__ x,
                             const float* __restrict__ w,
                             const float* __restrict__ b,
                             float* __restrict__ out) {
  size_t idx = (size_t)blockIdx.x * 256 + threadIdx.x;
  if (idx >= (size_t)NB * CC * F1 * T1) return;
  int t = (int)(idx & (T1 - 1));
  size_t cf = idx >> 10;
  int f = (int)(cf % F1);
  size_t nc = cf / F1;
  int c = (int)(nc & (CC - 1));
  int n = (int)(nc >> 5);
  float acc = b[c];
#pragma unroll
  for (int df = 0; df < 3; ++df) {
    int fi = 2 * f - 1 + df;
    if (fi < 0 || fi >= F0) continue;
#pragma unroll
    for (int dt = 0; dt < 3; ++dt) {
      int ti = 2 * t - 1 + dt;
      if (ti < 0 || ti >= T0) continue;
      acc += w[(c * 3 + df) * 3 + dt] * x[((size_t)n * F0 + fi) * T0 + ti];
    }
  }
  out[idx] = acc;
}

// ---------------------------------------------------------------------------
// LayerNorm over feature axis F (per n,c,t) + exact GELU.
// ---------------------------------------------------------------------------
__global__ void cnnln_gelu_kernel(const float* __restrict__ in,
                                  const float* __restrict__ g,
                                  const float* __restrict__ b,
                                  float* __restrict__ out) {
  int idx = blockIdx.x * 256 + threadIdx.x;   // 16*32*1024 columns
  if (idx >= NB * CC * T1) return;
  int t = idx & (T1 - 1);
  int c = (idx >> 10) & (CC - 1);
  int n = idx >> 15;
  const float* p = in + (((size_t)n * CC + c) * F1) * T1 + t;
  float s = 0.f, sq = 0.f;
  for (int f = 0; f < F1; ++f) {
    float v = p[(size_t)f * T1];
    s += v; sq += v * v;
  }
  float mean = s * (1.0f / F1);
  float var = sq * (1.0f / F1) - mean * mean;
  float rstd = rsqrtf(var + 1e-5f);
  float* q = out + (((size_t)n * CC + c) * F1) * T1 + t;
  for (int f = 0; f < F1; ++f) {
    float v = (p[(size_t)f * T1] - mean) * rstd * g[f] + b[f];
    q[(size_t)f * T1] = geluf(v);
  }
}

// ---------------------------------------------------------------------------
// 32->32 3x3 conv, pad 1, as implicit GEMM on WMMA f32 16x16x4.
// M = 32 out-channels (2 tiles), N = 64 time positions (4 tiles), K = 288.
// Weights async-staged into LDS (b128); haloed patch (32ch x 3 rows x 66t) and
// a 288-entry k->patch-offset LUT staged synchronously.
// One workgroup per (n, f, t-chunk).  128 threads = 4 waves, 2 tiles/wave.
// ---------------------------------------------------------------------------
template <bool ADDRES>
__global__ __launch_bounds__(128) void conv3x3_wmma_kernel(
    const float* __restrict__ in, const float* __restrict__ w,
    const float* __restrict__ bias, const float* __restrict__ resid,
    float* __restrict__ out) {
  __shared__ __align__(16) float wsm[CC * 288];        // 36864 B
  __shared__ __align__(16) float patch[CC * 3 * 66];   // 25344 B
  __shared__ int poff[288];                            //  1152 B
  const int t0 = blockIdx.x * 64;
  const int f = blockIdx.y;
  const int n = blockIdx.z;
  const int tid = threadIdx.x;

  // Async-stage the 32x288 weight panel: 2304 x b128, 18 per thread.
  for (int i = tid; i < (CC * 288) / 4; i += 128)
    async_copy16(&wsm[i * 4], &w[i * 4]);

  // k -> patch-offset LUT (pure function of k; removes div/mod from hot loop)
  for (int k = tid; k < 288; k += 128) {
    int ci = k / 9, rem = k - 9 * ci, df = rem / 3, dt = rem - 3 * df;
    poff[k] = (ci * 3 + df) * 66 + dt;
  }

  // Haloed input patch (has zero-padding -> synchronous staging).
  for (int i = tid; i < CC * 3 * 66; i += 128) {
    int tt = i % 66;
    int rest = i / 66;
    int df = rest % 3;
    int ci = rest / 3;
    int fi = f + df - 1;
    int ti = t0 - 1 + tt;
    float v = 0.f;
    if (fi >= 0 && fi < F1 && ti >= 0 && ti < T1)
      v = in[(((size_t)n * CC + ci) * F1 + fi) * T1 + ti];
    patch[i] = v;
  }
  wait_async0();
  __syncthreads();

  const int wave = tid >> 5, lane = tid & 31;
  const int laneLo = lane & 15, laneHi = lane >> 4;

  for (int i = 0; i < 2; ++i) {
    int tile = wave + 4 * i;        // 0..7
    int mt = tile & 1;              // co tile
    int jt = tile >> 1;             // t tile
    const int abase = (mt * 16 + laneLo) * 288;
    const int tt = jt * 16 + laneLo;
    v8f C = {};
#pragma unroll 4
    for (int kk = 0; kk < 72; ++kk) {
      int k = 4 * kk + 2 * laneHi;
      v2f A, B;
      A.x = wsm[abase + k];
      A.y = wsm[abase + k + 1];
      B.x = patch[poff[k] + tt];
      B.y = patch[poff[k + 1] + tt];
      C = wmma_f32(A, B, C);
    }
#pragma unroll
    for (int r = 0; r < 8; ++r) {
      int co = mt * 16 + r + 8 * laneHi;
      int t = t0 + jt * 16 + laneLo;
      size_t o = (((size_t)n * CC + co) * F1 + f) * T1 + t;
      float v = C[r] + bias[co];
      if (ADDRES) v += resid[o];
      out[o] = v;
    }
  }
}

// ---------------------------------------------------------------------------
// Fold BN1 into FC1: wp[d,c] = fc1_w[d,c]*g[c]*rsqrt(v[c]+eps)
//                    bias1[d] = sum_c fc1_w[d,c]*(b[c]-m[c]*g[c]*rsqrt(v+eps))
// ---------------------------------------------------------------------------
__global__ void fc1_prep_kernel(const float* __restrict__ fw,
                                const float* __restrict__ g,
                                const float* __restrict__ v,
                                float* __restrict__ wp) {
  int i = blockIdx.x * 256 + threadIdx.x;
  if (i >= RD * CF) return;
  int c = i % CF;
  wp[i] = fw[i] * g[c] * rsqrtf(v[c] + 1e-5f);
}

__global__ void fc1_bias_kernel(const float* __restrict__ fw,
                                const float* __restrict__ g,
                                const float* __restrict__ b,
                                const float* __restrict__ m,
                                const float* __restrict__ v,
                                float* __restrict__ bias1) {
  int d = threadIdx.x;  // 128
  float acc = 0.f;
  for (int c = 0; c < CF; ++c) {
    float s = g[c] * rsqrtf(v[c] + 1e-5f);
    acc += fw[d * CF + c] * (b[c] - m[c] * s);
  }
  bias1[d] = acc;
}

// ---------------------------------------------------------------------------
// FC1 WMMA GEMM:  seq[t,n,d] = bias1[d] + sum_c wp[d,c] * h[n,c,t]
// Per-workgroup: one n, 64 t's.  M(t) 4 tiles x N(d) 8 tiles, 8 waves x 4.
// ---------------------------------------------------------------------------
__global__ __launch_bounds__(256) void fc1_wmma_kernel(
    const float* __restrict__ h, const float* __restrict__ wp,
    const float* __restrict__ b1, float* __restrict__ seq) {
  const int t0 = blockIdx.x * 64;
  const int n = blockIdx.y;
  const int tid = threadIdx.x, wave = tid >> 5, lane = tid & 31;
  const int laneLo = lane & 15, laneHi = lane >> 4;
  for (int i = 0; i < 4; ++i) {
    int tile = wave + 8 * i;     // 0..31
    int mt = tile & 3;           // t tile
    int nt = tile >> 2;          // d tile
    int tb = t0 + mt * 16;
    int db = nt * 16;
    v8f C = {};
    for (int kk = 0; kk < CF / 4; ++kk) {
      int k = 4 * kk + 2 * laneHi;
      v2f A, B;
      A.x = h[((size_t)n * CF + k) * T1 + tb + laneLo];
      A.y = h[((size_t)n * CF + k + 1) * T1 + tb + laneLo];
      B.x = wp[(db + laneLo) * CF + k];
      B.y = wp[(db + laneLo) * CF + k + 1];
      C = wmma_f32(A, B, C);
    }
#pragma unroll
    for (int r = 0; r < 8; ++r) {
      int t = tb + r + 8 * laneHi;
      int d = db + laneLo;
      seq[((size_t)t * NB + n) * RD + d] = C[r] + b1[d];
    }
  }
}

// ---------------------------------------------------------------------------
// Batched input-gates GEMM: ig[row,g] = sum_k wih[g,k] * seq[row,k]
// rows = t*16+n (16384).  64 rows (32 KB) async-staged into LDS per workgroup.
// ---------------------------------------------------------------------------
__global__ __launch_bounds__(256) void igates_wmma_kernel(
    const float* __restrict__ seq, const float* __restrict__ wih,
    float* __restrict__ ig) {
  __shared__ __align__(16) float rows[64 * RD];   // 32 KB
  const int r0 = blockIdx.x * 64;
  const int gb = blockIdx.y;        // 0..3 -> g base 128*gb
  const int tid = threadIdx.x, wave = tid >> 5, lane = tid & 31;
  const int laneLo = lane & 15, laneHi = lane >> 4;
  // 2048 x b128 async copies, 8 per thread.
  const float* src = seq + (size_t)r0 * RD;
#pragma unroll
  for (int i = 0; i < 8; ++i) {
    int c4 = tid + i * 256;           // 16B chunk index
    async_copy16(&rows[c4 * 4], &src[c4 * 4]);
  }
  wait_async0();
  __syncthreads();
  for (int i = 0; i < 4; ++i) {
    int tile = wave + 8 * i;     // 0..31
    int mt = tile & 3;           // row tile
    int nt = tile >> 2;          // g tile 0..7
    int gbase = gb * 128 + nt * 16;
    v8f C = {};
#pragma unroll 4
    for (int kk = 0; kk < RD / 4; ++kk) {
      int k = 4 * kk + 2 * laneHi;
      v2f A, B;
      A.x = rows[(mt * 16 + laneLo) * RD + k];
      A.y = rows[(mt * 16 + laneLo) * RD + k + 1];
      B.x = wih[(gbase + laneLo) * RD + k];
      B.y = wih[(gbase + laneLo) * RD + k + 1];
      C = wmma_f32(A, B, C);
    }
#pragma unroll
    for (int r = 0; r < 8; ++r) {
      int row = r0 + mt * 16 + r + 8 * laneHi;
      ig[(size_t)row * GG + gbase + laneLo] = C[r];
    }
  }
}

// LN over rows of 512 (apply lnig/lnib) in place; one wave per row.
__global__ void ln_rows512_kernel(float* __restrict__ ig,
                                  const float* __restrict__ g,
                                  const float* __restrict__ b) {
  int row = blockIdx.x * 8 + (threadIdx.x >> 5);
  int lane = threadIdx.x & 31;
  if (row >= NROWS) return;
  float* p = ig + (size_t)row * GG;
  float s = 0.f, sq = 0.f;
  for (int i = lane; i < GG; i += 32) { float v = p[i]; s += v; sq += v * v; }
  for (int m = 1; m < 32; m <<= 1) {
    s += __shfl_xor(s, m, 32);
    sq += __shfl_xor(sq, m, 32);
  }
  float mean = s * (1.0f / GG);
  float var = sq * (1.0f / GG) - mean * mean;
  float rstd = rsqrtf(var + 1e-5f);
  for (int i = lane; i < GG; i += 32) p[i] = (p[i] - mean) * rstd * g[i] + b[i];
}

// ---------------------------------------------------------------------------
// Persistent LayerNorm-LSTM recurrence: ONE workgroup (32 waves) loops T steps.
// Wave w owns gate columns [16w,16w+16); its Whh B-fragments live in VGPRs for
// the whole kernel.  h, c, assembled gates live in LDS.  Per-step gate LN via
// LDS float atomics; igates for t+1 prefetched (global_prefetch_b8).
// ---------------------------------------------------------------------------
__global__ __launch_bounds__(1024) void lstm_rec_kernel(
    const float* __restrict__ ig,    // [T*16*512] (already LN'ed)
    const float* __restrict__ whh,   // [512,128]
    const float* __restrict__ lnhg, const float* __restrict__ lnhb,  // [512]
    const float* __restrict__ lncg, const float* __restrict__ lncb,  // [128]
    float* __restrict__ outSeq) {    // [T*16*128]
  __shared__ float hbuf[NB * HH];    // 8 KB
  __shared__ float cbuf[NB * HH];    // 8 KB
  __shared__ float gsm[NB * GG];     // 32 KB
  __shared__ float rsum[NB], rsq[NB];
  __shared__ float cmean[NB], crstd[NB];

  const int tid = threadIdx.x;
  const int wave = tid >> 5, lane = tid & 31;
  const int laneLo = lane & 15, laneHi = lane >> 4;
  const int g0 = wave * 16;

  // Register-resident Whh fragments: B[k][g]=whh[g][k], k = 4kk+2*laneHi+{0,1}
  v2f Bf[32];
#pragma unroll
  for (int kk = 0; kk < 32; ++kk) {
    int k = 4 * kk + 2 * laneHi;
    Bf[kk].x = whh[(g0 + laneLo) * HH + k];
    Bf[kk].y = whh[(g0 + laneLo) * HH + k + 1];
  }
  for (int i = tid; i < NB * HH; i += 1024) { hbuf[i] = 0.f; cbuf[i] = 0.f; }
  __syncthreads();

  const int e0 = tid, e1 = tid + 1024;   // each thread owns 2 of 2048 (n,j)

  for (int t = 0; t < T1; ++t) {
    if (tid < NB) { rsum[tid] = 0.f; rsq[tid] = 0.f; }
    if (t + 1 < T1 && tid < 256) {
      const float* p = ig + (size_t)(t + 1) * NB * GG;
      __builtin_prefetch(p + tid * 32, 0, 0);   // 256 x 128B = next 32KB block
    }
    __syncthreads();

    // hgates tile = h @ Whh^T  (A = h from LDS, B resident in VGPRs)
    v8f C = {};
#pragma unroll
    for (int kk = 0; kk < 32; ++kk) {
      int k = 4 * kk + 2 * laneHi;
      v2f A;
      A.x = hbuf[laneLo * HH + k];
      A.y = hbuf[laneLo * HH + k + 1];
      C = wmma_f32(A, Bf[kk], C);
    }
    // Cross-wave LN statistics over the 512-wide gate rows.
#pragma unroll
    for (int r = 0; r < 8; ++r) {
      float v = C[r];
      float s = v, sq = v * v;
      for (int m = 1; m < 16; m <<= 1) {
        s += __shfl_xor(s, m, 32);
        sq += __shfl_xor(sq, m, 32);
      }
      if (laneLo == 0) {
        int n = r + 8 * laneHi;
        atomicAdd(&rsum[n], s);
        atomicAdd(&rsq[n], sq);
      }
    }
    __syncthreads();

    // Normalize hgates, add precomputed igates, assemble into LDS.
    const float* igt = ig + (size_t)t * NB * GG;
#pragma unroll
    for (int r = 0; r < 8; ++r) {
      int n = r + 8 * laneHi;
      int g = g0 + laneLo;
      float mean = rsum[n] * (1.0f / GG);
      float var = rsq[n] * (1.0f / GG) - mean * mean;
      float x = (C[r] - mean) * rsqrtf(var + 1e-5f);
      gsm[n * GG + g] = x * lnhg[g] + lnhb[g] + igt[n * GG + g];
    }
    __syncthreads();

    // Cell update (each thread owns cbuf[e0], cbuf[e1] exclusively).
    float o0, o1, c0, c1;
    {
      int n = e0 >> 7, j = e0 & 127;
      float iv = gsm[n * GG + j], fv = gsm[n * GG + 128 + j];
      float gv = gsm[n * GG + 256 + j];
      o0 = gsm[n * GG + 384 + j];
      c0 = sigf(fv) * cbuf[e0] + sigf(iv) * tanhf(gv);
      cbuf[e0] = c0;
    }
    {
      int n = e1 >> 7, j = e1 & 127;
      float iv = gsm[n * GG + j], fv = gsm[n * GG + 128 + j];
      float gv = gsm[n * GG + 256 + j];
      o1 = gsm[n * GG + 384 + j];
      c1 = sigf(fv) * cbuf[e1] + sigf(iv) * tanhf(gv);
      cbuf[e1] = c1;
    }
    __syncthreads();

    // LN over c rows (128 wide): waves 0..15, one row each.
    if (wave < NB) {
      float s = 0.f, sq = 0.f;
      for (int j = lane; j < HH; j += 32) {
        float v = cbuf[wave * HH + j];
        s += v; sq += v * v;
      }
      for (int m = 1; m < 32; m <<= 1) {
        s += __shfl_xor(s, m, 32);
        sq += __shfl_xor(sq, m, 32);
      }
      if (lane == 0) {
        float mean = s * (1.0f / HH);
        float var = sq * (1.0f / HH) - mean * mean;
        cmean[wave] = mean;
        crstd[wave] = rsqrtf(var + 1e-5f);
      }
    }
    __syncthreads();

    float* outT = outSeq + (size_t)t * NB * HH;
    {
      int n = e0 >> 7, j = e0 & 127;
      float cl = (c0 - cmean[n]) * crstd[n] * lncg[j] + lncb[j];
      cbuf[e0] = cl;
      float h = sigf(o0) * tanhf(cl);
      hbuf[e0] = h;
      outT[e0] = h;
    }
    {
      int n = e1 >> 7, j = e1 & 127;
      float cl = (c1 - cmean[n]) * crstd[n] * lncg[j] + lncb[j];
      cbuf[e1] = cl;
      float h = sigf(o1) * tanhf(cl);
      hbuf[e1] = h;
      outT[e1] = h;
    }
    __syncthreads();
  }
}

// ---------------------------------------------------------------------------
// Head: y[row,cls] = sum_j fc2_w[cls,j] * BN2(seq[row,j])
// ---------------------------------------------------------------------------
__global__ void head_kernel(const float* __restrict__ seq,
                            const float* __restrict__ g,
                            const float* __restrict__ b,
                            const float* __restrict__ m,
                            const float* __restrict__ v,
                            const float* __restrict__ w2,
                            float* __restrict__ out) {
  int idx = blockIdx.x * 256 + threadIdx.x;
  if (idx >= NROWS * NCLS) return;
  int cls = idx % NCLS;
  int row = idx / NCLS;
  const float* p = seq + (size_t)row * HH;
  float acc = 0.f;
  for (int j = 0; j < HH; ++j) {
    float s = g[j] * rsqrtf(v[j] + 1e-5f);
    float x = (p[j] - m[j]) * s + b[j];
    acc += w2[cls * HH + j] * x;
  }
  out[idx] = acc;
}

// ---------------------------------------------------------------------------
extern "C" void kernel_launch(void* const* d_in, const int* in_sizes, int n_in,
                              void* d_out, int out_size, void* d_ws,
                              size_t ws_size, hipStream_t stream) {
  (void)in_sizes; (void)n_in; (void)out_size; (void)ws_size;
  const float* x       = (const float*)d_in[0];
  const float* conv0_w = (const float*)d_in[1];
  const float* conv0_b = (const float*)d_in[2];
  const float* rln1g   = (const float*)d_in[3];
  const float* rln1b   = (const float*)d_in[4];
  const float* rw1     = (const float*)d_in[5];
  const float* rb1     = (const float*)d_in[6];
  const float* rln2g   = (const float*)d_in[7];
  const float* rln2b   = (const float*)d_in[8];
  const float* rw2     = (const float*)d_in[9];
  const float* rb2     = (const float*)d_in[10];
  const float* bn1g    = (const float*)d_in[11];
  const float* bn1b    = (const float*)d_in[12];
  const float* bn1m    = (const float*)d_in[13];
  const float* bn1v    = (const float*)d_in[14];
  const float* fc1w    = (const float*)d_in[15];
  const float* wih     = (const float*)d_in[16];
  const float* whh     = (const float*)d_in[17];
  const float* lnig    = (const float*)d_in[18];
  const float* lnib    = (const float*)d_in[19];
  const float* lnhg    = (const float*)d_in[20];
  const float* lnhb    = (const float*)d_in[21];
  const float* lncg    = (const float*)d_in[22];
  const float* lncb    = (const float*)d_in[23];
  const float* bn2g    = (const float*)d_in[24];
  const float* bn2b    = (const float*)d_in[25];
  const float* bn2m    = (const float*)d_in[26];
  const float* bn2v    = (const float*)d_in[27];
  const float* fc2w    = (const float*)d_in[28];

  float* ws = (float*)d_ws;
  const size_t SZ_ACT = (size_t)NB * CC * F1 * T1;   // 34,078,720
  const size_t SZ_SEQ = (size_t)NROWS * RD;          //  2,097,152
  float* bufA  = ws;
  float* bufB  = bufA + SZ_ACT;
  float* bufC  = bufB + SZ_ACT;
  float* wp    = bufC + SZ_ACT;            // 128*2080
  float* bias1 = wp + (size_t)RD * CF;     // 128
  float* seqX  = bias1 + RD;
  float* seqY  = seqX + SZ_SEQ;
  float* gbuf  = seqY + SZ_SEQ;            // 16384*512

  // ---- front-end ----
  conv0_kernel<<<(int)((SZ_ACT + 255) / 256), 256, 0, stream>>>(
      x, conv0_w, conv0_b, bufA);

  for (int l = 0; l < 3; ++l) {
    cnnln_gelu_kernel<<<(NB * CC * T1) / 256, 256, 0, stream>>>(
        bufA, rln1g + l * F1, rln1b + l * F1, bufB);
    conv3x3_wmma_kernel<false><<<dim3(16, F1, NB), 128, 0, stream>>>(
        bufB, rw1 + (size_t)l * CC * CC * 9, rb1 + l * CC, nullptr, bufC);
    cnnln_gelu_kernel<<<(NB * CC * T1) / 256, 256, 0, stream>>>(
        bufC, rln2g + l * F1, rln2b + l * F1, bufB);
    conv3x3_wmma_kernel<true><<<dim3(16, F1, NB), 128, 0, stream>>>(
        bufB, rw2 + (size_t)l * CC * CC * 9, rb2 + l * CC, bufA, bufA);
  }

  // ---- BN1 folded into FC1, then WMMA GEMM into [T,N,D] ----
  fc1_prep_kernel<<<(RD * CF + 255) / 256, 256, 0, stream>>>(fc1w, bn1g, bn1v, wp);
  fc1_bias_kernel<<<1, RD, 0, stream>>>(fc1w, bn1g, bn1b, bn1m, bn1v, bias1);
  fc1_wmma_kernel<<<dim3(16, NB), 256, 0, stream>>>(bufA, wp, bias1, seqX);

  // ---- 5x LayerNorm-LSTM ----
  float* sin = seqX;
  float* sout = seqY;
  for (int l = 0; l < 5; ++l) {
    igates_wmma_kernel<<<dim3(NROWS / 64, 4), 256, 0, stream>>>(
        sin, wih + (size_t)l * GG * RD, gbuf);
    ln_rows512_kernel<<<NROWS / 8, 256, 0, stream>>>(
        gbuf, lnig + l * GG, lnib + l * GG);
    lstm_rec_kernel<<<1, 1024, 0, stream>>>(
        gbuf, whh + (size_t)l * GG * HH, lnhg + l * GG, lnhb + l * GG,
        lncg + l * HH, lncb + l * HH, sout);
    float* tmp = sin; sin = sout; sout = tmp;
  }

  // ---- BN2 + FC2 head ----
  head_kernel<<<(NROWS * NCLS + 255) / 256, 256, 0, stream>>>(
      sin, bn2g, bn2b, bn2m, bn2v, fc2w, (float*)d_out);
}